// GaussianTransformLayerC3_43946105372848
// MI455X (gfx1250) — compile-verified
//
#include <hip/hip_runtime.h>
#include <math.h>

typedef float v2f   __attribute__((ext_vector_type(2)));
typedef float v8f   __attribute__((ext_vector_type(8)));
typedef float f32x4 __attribute__((ext_vector_type(4)));

namespace {
constexpr int S_LEN = 4096;
constexpr int CH    = 64;
constexpr int SB    = 64;     // output s-rows per block
constexpr int HALO  = 31;     // left halo (max shift)
constexpr int ROWS  = 128;    // LDS tile rows: sB-31 .. sB+96
constexpr int LSTR  = 80;     // padded LDS row stride (floats) -> conflict-free B gathers

// scale 0: a=1, f len 11, H len 12, shift 6 , KPAD 28, hp base 0  , hp len 43 (d=-15..27)
// scale 1: a=3, f len 31, H len 32, shift 16, KPAD 48, hp base 43 , hp len 63 (d=-15..47)
// scale 2: a=6, f len 61, H len 62, shift 31, KPAD 80, hp base 106, hp len 95 (d=-15..79)
constexpr int F_TOTAL  = 11 + 31 + 61;   // 103
constexpr int HP_TOTAL = 43 + 63 + 95;   // 201
}

// Implicit-GEMM inner loop: D(16s x 16c) += Toeplitz(H) * X-slab, two s-tiles per call
// sharing each A fragment.  f32 WMMA layouts (wave32):
//   A 16x4 : lane<16 -> M=lane, {K0,K1}; lane>=16 -> M=lane-16, {K2,K3}
//   B 4x16 : lane<16 -> N=lane, {K0,K1}; lane>=16 -> N=lane-16, {K2,K3}
template<int KPAD, int SHIFT, int HPBASE>
__device__ __forceinline__ void conv_pair(const float* tile, const float* hp,
                                          int lane, int st0, int st1, int c0,
                                          v8f& acc0, v8f& acc1)
{
  const int i   = lane & 15;
  const int hi  = lane >> 4;
  const int rb0 = (HALO - SHIFT + 16 * st0) * LSTR + c0 + i;
  const int rb1 = (HALO - SHIFT + 16 * st1) * LSTR + c0 + i;
  const int ab  = HPBASE + 15 - i;        // zero-padded Toeplitz gather base
#pragma unroll
  for (int k0 = 0; k0 < KPAD; k0 += 4) {
    const int t0 = k0 + 2 * hi;
    v2f a, b0, b1;
    a.x  = hp[ab + t0];
    a.y  = hp[ab + t0 + 1];
    b0.x = tile[rb0 +  t0      * LSTR];
    b0.y = tile[rb0 + (t0 + 1) * LSTR];
    b1.x = tile[rb1 +  t0      * LSTR];
    b1.y = tile[rb1 + (t0 + 1) * LSTR];
    acc0 = __builtin_amdgcn_wmma_f32_16x16x4_f32(false, a, false, b0, (short)0, acc0, false, false);
    acc1 = __builtin_amdgcn_wmma_f32_16x16x4_f32(false, a, false, b1, (short)0, acc1, false, false);
  }
}

// D layout: VGPR r -> M = r + 8*(lane>=16), N = lane%16
__device__ __forceinline__ void store_tile(float* __restrict__ out, int b, int sc,
                                           int row0, int c0, int lane, v8f acc)
{
  const int j  = lane & 15;
  const int hi = lane >> 4;
  const size_t base = (((size_t)b * 3 + sc) * S_LEN + (size_t)(row0 + 8 * hi)) * CH
                      + (size_t)(c0 + j);
#pragma unroll
  for (int r = 0; r < 8; ++r)
    out[base + (size_t)r * CH] = acc[r];
}

__global__ void __launch_bounds__(256)
cwt_gaus1_wmma_kernel(const float* __restrict__ x, float* __restrict__ out)
{
  __shared__ float  tile[ROWS * LSTR];   // 40 KB signal tile
  __shared__ double psum[256];           // cumsum partials
  __shared__ float  fbuf[F_TOTAL];       // int_psi[j] samples (pywt filters)
  __shared__ float  hp[HP_TOTAL];        // zero-padded derivative kernels * -sqrt(a)

  const int tid = threadIdx.x;
  const int b   = blockIdx.x >> 6;
  const int sB  = (blockIdx.x & 63) * SB;

  const double step = 10.0 / 1023.0;
  const double invC = 1.0 / pow(M_PI * 0.5, 0.25);   // 1/(pi/2)^0.25

  // ---- build pywt gaus1 integrated-wavelet filters (once per block) ----
  {
    double s = 0.0;
#pragma unroll
    for (int q = 0; q < 4; ++q) {
      const int    i  = tid * 4 + q;                 // 1024 psi samples
      const double xv = -5.0 + step * (double)i;
      s += -2.0 * xv * exp(-xv * xv) * invC;
    }
    psum[tid] = s;
  }
  __syncthreads();
  if (tid == 0) {                                    // exclusive prefix over 256 partials
    double run = 0.0;
    for (int t = 0; t < 256; ++t) { const double tmp = psum[t]; psum[t] = run; run += tmp; }
  }
  __syncthreads();
  if (tid < F_TOTAL) {                               // f_a[k] = int_psi[trunc(k/(a*step))]
    int a, k;
    if (tid < 11)      { a = 1; k = tid;      }
    else if (tid < 42) { a = 3; k = tid - 11; }
    else               { a = 6; k = tid - 42; }
    const long j = (long)((double)k / ((double)a * step));
    double v = psum[j >> 2];
    for (long i = j & ~3L; i <= j; ++i) {
      const double xv = -5.0 + step * (double)i;
      v += -2.0 * xv * exp(-xv * xv) * invC;
    }
    fbuf[tid] = (float)(v * step);
  }
  __syncthreads();
  if (tid < HP_TOTAL) {                              // H_a[m] = -sqrt(a)*(f[m-1]-f[m]), zero-padded
    int d, foff, LH; double aa;
    if (tid < 43)       { d = tid - 15;       foff = 0;  LH = 12; aa = 1.0; }
    else if (tid < 106) { d = tid - 43 - 15;  foff = 11; LH = 32; aa = 3.0; }
    else                { d = tid - 106 - 15; foff = 42; LH = 62; aa = 6.0; }
    float v = 0.0f;
    if (d >= 0 && d < LH) {
      const float fm1 = (d >= 1)      ? fbuf[foff + d - 1] : 0.0f;
      const float fm  = (d < LH - 1)  ? fbuf[foff + d]     : 0.0f;
      v = (float)(-sqrt(aa)) * (fm1 - fm);
    }
    hp[tid] = v;
  }

  // ---- load 128x64 fp32 signal tile (zero-padded halo), float4 coalesced ----
#pragma unroll
  for (int it = 0; it < 8; ++it) {
    const int e4  = tid + it * 256;        // float4 index 0..2047
    const int row = e4 >> 4;
    const int col = (e4 & 15) << 2;
    const int gr  = sB - HALO + row;
    f32x4 v = {0.0f, 0.0f, 0.0f, 0.0f};
    if (gr >= 0 && gr < S_LEN)
      v = *(const f32x4*)(x + ((size_t)b * S_LEN + (size_t)gr) * CH + col);
    *(f32x4*)(&tile[row * LSTR + col]) = v;
  }
  __syncthreads();

  // ---- wave -> (c-tile, pair of s-tiles); 3 scales each ----
  const int lane = tid & 31;
  const int w    = tid >> 5;          // 0..7
  const int c0   = (w & 3) * 16;
  const int st0  = (w >> 2) * 2;
  const int st1  = st0 + 1;

  const v8f z = {0.f, 0.f, 0.f, 0.f, 0.f, 0.f, 0.f, 0.f};

  {
    v8f a0 = z, a1 = z;
    conv_pair<28, 6, 0>(tile, hp, lane, st0, st1, c0, a0, a1);
    store_tile(out, b, 0, sB + 16 * st0, c0, lane, a0);
    store_tile(out, b, 0, sB + 16 * st1, c0, lane, a1);
  }
  {
    v8f a0 = z, a1 = z;
    conv_pair<48, 16, 43>(tile, hp, lane, st0, st1, c0, a0, a1);
    store_tile(out, b, 1, sB + 16 * st0, c0, lane, a0);
    store_tile(out, b, 1, sB + 16 * st1, c0, lane, a1);
  }
  {
    v8f a0 = z, a1 = z;
    conv_pair<80, 31, 106>(tile, hp, lane, st0, st1, c0, a0, a1);
    store_tile(out, b, 2, sB + 16 * st0, c0, lane, a0);
    store_tile(out, b, 2, sB + 16 * st1, c0, lane, a1);
  }
}

extern "C" void kernel_launch(void* const* d_in, const int* in_sizes, int n_in,
                              void* d_out, int out_size, void* d_ws, size_t ws_size,
                              hipStream_t stream) {
  (void)in_sizes; (void)n_in; (void)out_size; (void)d_ws; (void)ws_size;
  const float* x  = (const float*)d_in[0];
  float*      out = (float*)d_out;
  // 32 batches * 64 s-slabs of 64 rows
  hipLaunchKernelGGL(cwt_gaus1_wmma_kernel, dim3(32 * 64), dim3(256), 0, stream, x, out);
}